// GCN_21638045237575
// MI455X (gfx1250) — compile-verified
//
#include <hip/hip_runtime.h>

typedef float v2f __attribute__((ext_vector_type(2)));
typedef float v8f __attribute__((ext_vector_type(8)));

#define EPS 1e-5f
#define SLOPE 0.01f

// ---------------------------------------------------------------------------
// Degree / normalization
// ---------------------------------------------------------------------------
__global__ void deg_init_kernel(float* __restrict__ deg, int n) {
    int i = blockIdx.x * blockDim.x + threadIdx.x;
    if (i < n) deg[i] = 1.0f;   // self-loop contributes 1
}

__global__ void deg_accum_kernel(const int* __restrict__ dst, float* __restrict__ deg, int e) {
    int i = blockIdx.x * blockDim.x + threadIdx.x;
    if (i < e) atomicAdd(&deg[dst[i]], 1.0f);
}

__global__ void deg_rsqrt_kernel(float* __restrict__ deg, int n) {
    int i = blockIdx.x * blockDim.x + threadIdx.x;
    if (i < n) deg[i] = rsqrtf(deg[i]);   // deg buffer becomes dis = deg^{-1/2}
}

// ---------------------------------------------------------------------------
// FP32 WMMA GEMM: C[nrows, DOUT] = A[nrows, DIN] @ W[DIN, DOUT]
// One wave computes one 16x16 output tile via V_WMMA_F32_16X16X4_F32.
// A 16x4 layout: lane row = lane&15, K = 2*(lane>>4) + vgpr
// B 4x16 layout: lane col = lane&15, K = 2*(lane>>4) + vgpr
// C/D: VGPR v -> row = v + 8*(lane>>4), col = lane&15
// K-tail padding (DIN=6) is branch-free: clamped-index loads * {0,1} mask,
// so EXEC stays all-1s through the whole WMMA stream.
// Requires nrows % 16 == 0 (true here: 100000 = 6250*16).
// ---------------------------------------------------------------------------
template <int DIN, int DOUT>
__global__ void gemm_wmma_f32_kernel(const float* __restrict__ A,
                                     const float* __restrict__ W,
                                     float* __restrict__ C, int nrows) {
    constexpr int TILES_N = DOUT / 16;
    int wave = (blockIdx.x * blockDim.x + threadIdx.x) >> 5;
    int lane = threadIdx.x & 31;
    int tile_m = wave / TILES_N;
    int row0 = tile_m << 4;
    if (row0 >= nrows) return;               // wave-uniform exit: EXEC stays all-1s
    int col0 = (wave - tile_m * TILES_N) << 4;

    int n    = lane & 15;
    int half = lane >> 4;

    const float* Arow = A + (size_t)(row0 + n) * DIN;
    const float* Wcol = W + col0 + n;
    v8f acc = {0.f, 0.f, 0.f, 0.f, 0.f, 0.f, 0.f, 0.f};

#pragma unroll
    for (int k0 = 0; k0 < DIN; k0 += 4) {
        int ka = k0 + (half << 1);
        // branch-free zero padding for the K tail (only materializes for DIN=6)
        float m0 = (ka     < DIN) ? 1.0f : 0.0f;
        float m1 = (ka + 1 < DIN) ? 1.0f : 0.0f;
        int   i0 = (ka     < DIN) ? ka     : 0;
        int   i1 = (ka + 1 < DIN) ? ka + 1 : 0;
        v2f a, b;
        a.x = Arow[i0] * m0;
        a.y = Arow[i1] * m1;
        b.x = Wcol[(size_t)i0 * DOUT] * m0;
        b.y = Wcol[(size_t)i1 * DOUT] * m1;
        acc = __builtin_amdgcn_wmma_f32_16x16x4_f32(
            /*neg_a=*/false, a, /*neg_b=*/false, b,
            /*c_mod=*/(short)0, acc, /*reuse_a=*/false, /*reuse_b=*/false);
    }

    float* Crow = C + (size_t)row0 * DOUT + col0 + n;
#pragma unroll
    for (int v = 0; v < 8; ++v)
        Crow[(size_t)(v + (half << 3)) * DOUT] = acc[v];
}

// ---------------------------------------------------------------------------
// agg = bias + hW * dis[i]^2   (bias + self-loop term; initializes agg buffer)
// ---------------------------------------------------------------------------
template <int DOUT>
__global__ void init_agg_kernel(const float* __restrict__ hW,
                                const float* __restrict__ bias,
                                const float* __restrict__ dis,
                                float* __restrict__ agg, int n) {
    int i = blockIdx.x * blockDim.x + threadIdx.x;
    if (i >= n * DOUT) return;
    int row = i / DOUT;          // power of two -> shift
    int c   = i & (DOUT - 1);
    float d = dis[row];
    agg[i] = bias[c] + hW[i] * d * d;
}

// ---------------------------------------------------------------------------
// Edge scatter: agg[dst] += hW[src] * (dis[src]*dis[dst])
// Thread handles one edge x 4 channels (float4 gather, 4 global f32 atomics).
// Adjacent threads -> adjacent channel groups of the same edge row.
// All addressing is shift/mask (GROUPS is a power of two).
// ---------------------------------------------------------------------------
template <int DOUT>
__global__ void scatter_kernel(const float* __restrict__ hW,
                               const int* __restrict__ src,
                               const int* __restrict__ dst,
                               const float* __restrict__ dis,
                               float* __restrict__ agg, int e) {
    constexpr int GROUPS = DOUT >> 2;
    long long tid = (long long)blockIdx.x * blockDim.x + threadIdx.x;
    if (tid >= (long long)e * GROUPS) return;
    int edge = (int)(tid / GROUPS);               // power of two -> shift
    int c4   = ((int)tid & (GROUPS - 1)) << 2;
    int s = src[edge];
    int d = dst[edge];
    float w = dis[s] * dis[d];
    float4 v = *(const float4*)(hW + (size_t)s * DOUT + c4);
    float* ap = agg + (size_t)d * DOUT + c4;
    atomicAdd(ap + 0, v.x * w);
    atomicAdd(ap + 1, v.y * w);
    atomicAdd(ap + 2, v.z * w);
    atomicAdd(ap + 3, v.w * w);
}

// ---------------------------------------------------------------------------
// BN statistics: one block per channel; biased variance (training-mode BN).
// Emits scale = gamma * rsqrt(var+eps), shift = beta - mean*scale.
// ---------------------------------------------------------------------------
__global__ void bn_stats_kernel(const float* __restrict__ agg, int n, int dout,
                                const float* __restrict__ gamma,
                                const float* __restrict__ beta,
                                float* __restrict__ scale,
                                float* __restrict__ shift) {
    int c = blockIdx.x;
    float s = 0.f, ss = 0.f;
    for (int i = threadIdx.x; i < n; i += blockDim.x) {
        float v = agg[(size_t)i * dout + c];
        s += v;
        ss += v * v;
    }
    __shared__ float red0[256];
    __shared__ float red1[256];
    red0[threadIdx.x] = s;
    red1[threadIdx.x] = ss;
    __syncthreads();
    for (int off = 128; off > 0; off >>= 1) {
        if ((int)threadIdx.x < off) {
            red0[threadIdx.x] += red0[threadIdx.x + off];
            red1[threadIdx.x] += red1[threadIdx.x + off];
        }
        __syncthreads();
    }
    if (threadIdx.x == 0) {
        float inv_n = 1.0f / (float)n;
        float mean = red0[0] * inv_n;
        float var = red1[0] * inv_n - mean * mean;
        float sc = gamma[c] * rsqrtf(var + EPS);
        scale[c] = sc;
        shift[c] = beta[c] - mean * sc;
    }
}

// ---------------------------------------------------------------------------
// Fused BN-apply + LeakyReLU, in place.
// ---------------------------------------------------------------------------
template <int DOUT>
__global__ void bn_lrelu_kernel(float* __restrict__ h,
                                const float* __restrict__ scale,
                                const float* __restrict__ shift, int n) {
    int i = blockIdx.x * blockDim.x + threadIdx.x;
    if (i >= n * DOUT) return;
    int c = i & (DOUT - 1);
    float v = h[i] * scale[c] + shift[c];
    h[i] = (v >= 0.f) ? v : SLOPE * v;
}

// ---------------------------------------------------------------------------
// Final linear: out[N,3] = h[N,16] @ W_out[16,3] + b_out
// ---------------------------------------------------------------------------
__global__ void out_linear_kernel(const float* __restrict__ h,
                                  const float* __restrict__ Wout,
                                  const float* __restrict__ bout,
                                  float* __restrict__ out, int n) {
    int i = blockIdx.x * blockDim.x + threadIdx.x;
    if (i >= n) return;
    float a0 = bout[0], a1 = bout[1], a2 = bout[2];
    const float* hr = h + (size_t)i * 16;
#pragma unroll
    for (int k = 0; k < 16; ++k) {
        float v = hr[k];
        a0 += v * Wout[k * 3 + 0];
        a1 += v * Wout[k * 3 + 1];
        a2 += v * Wout[k * 3 + 2];
    }
    out[i * 3 + 0] = a0;
    out[i * 3 + 1] = a1;
    out[i * 3 + 2] = a2;
}

// ---------------------------------------------------------------------------
// Host-side layer driver
// ---------------------------------------------------------------------------
template <int DIN, int DOUT>
static void run_layer(const float* hin, const float* W, const float* b,
                      const float* gamma, const float* beta,
                      const float* dis, const int* src, const int* dst,
                      float* hW, float* agg, float* scale, float* shift,
                      int N, int E, hipStream_t stream) {
    // GEMM: hW = hin @ W
    int tiles = ((N + 15) / 16) * (DOUT / 16);
    int blocks = (tiles + 7) / 8;   // 8 waves per 256-thread block
    gemm_wmma_f32_kernel<DIN, DOUT><<<blocks, 256, 0, stream>>>(hin, W, hW, N);

    // agg = bias + self-loop term
    int total = N * DOUT;
    init_agg_kernel<DOUT><<<(total + 255) / 256, 256, 0, stream>>>(hW, b, dis, agg, N);

    // edge scatter-add
    long long etotal = (long long)E * (DOUT / 4);
    int eblocks = (int)((etotal + 255) / 256);
    scatter_kernel<DOUT><<<eblocks, 256, 0, stream>>>(hW, src, dst, dis, agg, E);

    // BN stats + fused BN/LeakyReLU apply (in place on agg)
    bn_stats_kernel<<<DOUT, 256, 0, stream>>>(agg, N, DOUT, gamma, beta, scale, shift);
    bn_lrelu_kernel<DOUT><<<(total + 255) / 256, 256, 0, stream>>>(agg, scale, shift, N);
}

extern "C" void kernel_launch(void* const* d_in, const int* in_sizes, int n_in,
                              void* d_out, int out_size, void* d_ws, size_t ws_size,
                              hipStream_t stream) {
    (void)n_in; (void)out_size; (void)ws_size;

    const float* x    = (const float*)d_in[0];
    const int*   eidx = (const int*)d_in[1];
    int N = in_sizes[0] / 6;
    int E = in_sizes[1] / 2;
    const int* src = eidx;        // edge_index[0]
    const int* dst = eidx + E;    // edge_index[1]

    const float* W[5];
    const float* b[5];
    const float* gm[5];
    const float* bt[5];
    for (int l = 0; l < 5; ++l) {
        W[l]  = (const float*)d_in[2 + 4 * l + 0];
        b[l]  = (const float*)d_in[2 + 4 * l + 1];
        gm[l] = (const float*)d_in[2 + 4 * l + 2];
        bt[l] = (const float*)d_in[2 + 4 * l + 3];
    }
    const float* Wout = (const float*)d_in[22];
    const float* bout = (const float*)d_in[23];

    // Workspace layout (bytes):
    //  [0, 4N)                      dis (also used as deg during setup)
    //  [4N, 4N + 512N)              bufA  (N x 128 floats)  -- agg / layer input
    //  [4N+512N, 4N + 1024N)        bufB  (N x 128 floats)  -- GEMM output hW
    //  then scale[128], shift[128]
    char* ws = (char*)d_ws;
    float* dis  = (float*)ws;
    float* bufA = (float*)(ws + (size_t)4 * N);
    float* bufB = (float*)(ws + (size_t)4 * N + (size_t)512 * N);
    float* scale = (float*)(ws + (size_t)4 * N + (size_t)1024 * N);
    float* shift = scale + 128;

    // --- degree + symmetric normalization ---
    deg_init_kernel<<<(N + 255) / 256, 256, 0, stream>>>(dis, N);
    deg_accum_kernel<<<(E + 255) / 256, 256, 0, stream>>>(dst, dis, E);
    deg_rsqrt_kernel<<<(N + 255) / 256, 256, 0, stream>>>(dis, N);

    // --- 5 GCN layers; bufA/bufB ping-pong: GEMM reads hin, writes bufB;
    //     scatter reads bufB, writes bufA; BN in place on bufA ---
    run_layer<6, 32>   (x,    W[0], b[0], gm[0], bt[0], dis, src, dst, bufB, bufA, scale, shift, N, E, stream);
    run_layer<32, 128> (bufA, W[1], b[1], gm[1], bt[1], dis, src, dst, bufB, bufA, scale, shift, N, E, stream);
    run_layer<128, 128>(bufA, W[2], b[2], gm[2], bt[2], dis, src, dst, bufB, bufA, scale, shift, N, E, stream);
    run_layer<128, 32> (bufA, W[3], b[3], gm[3], bt[3], dis, src, dst, bufB, bufA, scale, shift, N, E, stream);
    run_layer<32, 16>  (bufA, W[4], b[4], gm[4], bt[4], dis, src, dst, bufB, bufA, scale, shift, N, E, stream);

    // --- final linear 16 -> 3 ---
    out_linear_kernel<<<(N + 255) / 256, 256, 0, stream>>>(bufA, Wout, bout, (float*)d_out, N);
}